// TemporalFeatureMapAttention_27101243637923
// MI455X (gfx1250) — compile-verified
//
#include <hip/hip_runtime.h>
#include <stdint.h>

// MI455X (gfx1250) fused temporal feature-map attention.
//
// out[b,c,hw] = seq[4,b,c,hw] + gamma * sum_t softmax_t(score[t,b,hw]) * seq[t,b,c,hw]
// score[t] = sum_c seq[t,c]*w_other[c]  (+ center score + bias, which are constant
// over t and cancel inside the softmax -> dropped).
//
// Single pass over seq (231 MB) via online softmax with t=4 processed last so the
// center frame is still resident in LDS for the epilogue. Frames are staged
// global->LDS with double-buffered async DMA (global_load_async_to_lds_b128,
// ASYNCcnt). Score dot-products run on v_wmma_f32_16x16x4_f32.

typedef __attribute__((ext_vector_type(2))) float v2f;
typedef __attribute__((ext_vector_type(8))) float v8f;

namespace {
constexpr int kT = 9, kB = 8, kC = 256, kHW = 56 * 56;   // 3136
constexpr int kPX = 32;                                   // pixels per workgroup
constexpr int kTilesPerB = kHW / kPX;                     // 98
constexpr int kTileDW = kC * kPX;                         // 8192 dwords = 32 KB / frame tile
constexpr uint32_t kVbufBytes = (uint32_t)kTileDW * 4u;   // 32768
constexpr uint32_t kWbufOff = 2u * kVbufBytes;            // 65536 : w_other[256]
constexpr uint32_t kSbufOff = kWbufOff + (uint32_t)kC * 4u;     // 66560 : 4x32 partial scores
constexpr uint32_t kSmemBytes = kSbufOff + 4u * (uint32_t)kPX * 4u;  // 67072
}

__device__ __forceinline__ void async_ld_b128(uint32_t lds_byte, const void* gaddr) {
  asm volatile("global_load_async_to_lds_b128 %0, %1, off"
               :: "v"(lds_byte), "v"((uint64_t)(uintptr_t)gaddr)
               : "memory");
}
__device__ __forceinline__ void async_ld_b32(uint32_t lds_byte, const void* gaddr) {
  asm volatile("global_load_async_to_lds_b32 %0, %1, off"
               :: "v"(lds_byte), "v"((uint64_t)(uintptr_t)gaddr)
               : "memory");
}
__device__ __forceinline__ void wait_async_zero() {
  asm volatile("s_wait_asynccnt 0x0" ::: "memory");
}

__global__ __launch_bounds__(256)
void tfma_fused_kernel(const float* __restrict__ seq,
                       const float* __restrict__ w,
                       const float* __restrict__ gamma_p,
                       float* __restrict__ out) {
  extern __shared__ float smem[];
  const int tid  = threadIdx.x;
  const int wid  = tid >> 5;     // 8 waves
  const int lane = tid & 31;

  const int tile = blockIdx.x;
  const int b    = tile / kTilesPerB;
  const int hw0  = (tile - b * kTilesPerB) * kPX;

  const uint32_t smem_base = (uint32_t)(uintptr_t)smem;
  float* const wbuf = smem + (kWbufOff >> 2);
  float* const sbuf = smem + (kSbufOff >> 2);

  // One-time: stage w_other = w[0, C:2C, 0, 0] into LDS via async DMA.
  async_ld_b32(smem_base + kWbufOff + (uint32_t)tid * 4u, w + kC + tid);

  const int t_order[kT] = {0, 1, 2, 3, 5, 6, 7, 8, 4};   // center frame last

  auto issue_tile = [&](int t, int parity) {
    const size_t base = (size_t)(t * kB + b) * kC * kHW;
    const uint32_t lds0 = smem_base + (uint32_t)parity * kVbufBytes;
#pragma unroll
    for (int k = 0; k < 8; ++k) {                 // 8 x 16B chunks per thread
      const int flat4 = k * 256 + tid;            // 16-byte chunk id in tile
      const int c  = flat4 >> 3;                  // 8 chunks per 32-px channel row
      const int jj = (flat4 & 7) << 2;            // dword offset within row
      async_ld_b128(lds0 + (uint32_t)flat4 * 16u,
                    seq + base + (size_t)c * kHW + (size_t)(hw0 + jj));
    }
  };

  issue_tile(t_order[0], 0);

  const float gmm = gamma_p[0];

  // Online-softmax state: each thread owns pixel `px` and channels [c0, c0+32).
  float m = -3.0e38f, l = 0.0f;
  float acc[32];
#pragma unroll
  for (int j = 0; j < 32; ++j) acc[j] = 0.0f;

  // WMMA task split: wave -> (16-px group, 64-channel block)
  const int pxg   = wid & 1;
  const int chalf = wid >> 1;
  const int px0   = pxg * 16;
  const int cbase = chalf * 64;
  const int half  = lane >> 4;    // lane half selects K pair (16x4 f32 A/B layout)
  const int lx    = lane & 15;
  const int px    = lane;         // update phase: wave covers all 32 pixels
  const int c0    = wid * 32;     // update phase: wave owns a 32-channel slab

  for (int i = 0; i < kT; ++i) {
    wait_async_zero();
    __syncthreads();               // current frame tile resident + prev buffer free
    const int p = i & 1;
    if (i + 1 < kT) issue_tile(t_order[i + 1], (i + 1) & 1);

    const float* vb = smem + p * kTileDW;

    // ---- scores: 16px x 64c partial dot per wave via v_wmma_f32_16x16x4_f32.
    // A[m][k] = w_other[c+k] (all rows equal); B[k][n] = v[c+k][px0+n].
    {
      v8f d0 = {0, 0, 0, 0, 0, 0, 0, 0};
      v8f d1 = {0, 0, 0, 0, 0, 0, 0, 0};
#pragma unroll
      for (int step = 0; step < 8; ++step) {
        const int ca = cbase + step * 4 + half * 2;
        const int cb = cbase + 32 + step * 4 + half * 2;
        v2f a0; a0[0] = wbuf[ca]; a0[1] = wbuf[ca + 1];
        v2f b0; b0[0] = vb[ca * kPX + px0 + lx];
                b0[1] = vb[(ca + 1) * kPX + px0 + lx];
        d0 = __builtin_amdgcn_wmma_f32_16x16x4_f32(false, a0, false, b0,
                                                   (short)0, d0, false, false);
        v2f a1; a1[0] = wbuf[cb]; a1[1] = wbuf[cb + 1];
        v2f b1; b1[0] = vb[cb * kPX + px0 + lx];
                b1[1] = vb[(cb + 1) * kPX + px0 + lx];
        d1 = __builtin_amdgcn_wmma_f32_16x16x4_f32(false, a1, false, b1,
                                                   (short)0, d1, false, false);
      }
      const float spart = d0[0] + d1[0];   // row 0 (== every row) of D
      if (lane < 16) sbuf[chalf * 32 + px0 + lane] = spart;
    }
    __syncthreads();

    // ---- online softmax update (thread-redundant m,l per pixel; disjoint acc)
    const float s  = sbuf[px] + sbuf[32 + px] + sbuf[64 + px] + sbuf[96 + px];
    const float m2 = fmaxf(m, s);
    const float fo = __expf(m - m2);
    const float pe = __expf(s - m2);
    l = l * fo + pe;
    m = m2;
#pragma unroll
    for (int j = 0; j < 32; ++j) {
      const float v = vb[(c0 + j) * kPX + px];
      acc[j] = acc[j] * fo + pe * v;
    }

    if (i == kT - 1) {
      // t==4 (center) tile is still resident in vb -> epilogue without re-read.
      const float gl = gmm / l;
      const size_t ob = (size_t)b * kC * kHW;
#pragma unroll
      for (int j = 0; j < 32; ++j) {
        const float cv = vb[(c0 + j) * kPX + px];
        out[ob + (size_t)(c0 + j) * kHW + (size_t)(hw0 + px)] = fmaf(acc[j], gl, cv);
      }
    }
  }
}

extern "C" void kernel_launch(void* const* d_in, const int* in_sizes, int n_in,
                              void* d_out, int out_size, void* d_ws, size_t ws_size,
                              hipStream_t stream) {
  (void)in_sizes; (void)n_in; (void)d_ws; (void)ws_size; (void)out_size;
  const float* seq   = (const float*)d_in[0];
  const float* w     = (const float*)d_in[1];
  // d_in[2] = bias: constant across t, cancels inside the softmax -> unused.
  const float* gamma = (const float*)d_in[3];
  float* out = (float*)d_out;

  dim3 grid(kB * kTilesPerB);   // 784 workgroups
  dim3 block(256);              // 8 wave32s
  tfma_fused_kernel<<<grid, block, kSmemBytes, stream>>>(seq, w, gamma, out);
}